// SequentialGeometricUpdate_15857019256954
// MI455X (gfx1250) — compile-verified
//
#include <hip/hip_runtime.h>
#include <math.h>

// Problem constants (match reference)
#define KNB  16      // k nearest neighbors
#define NPTS 4096    // points per batch
#define MFPS 512     // FPS samples
#define EPSN 1e-12f

typedef __attribute__((ext_vector_type(2))) float v2f;  // A/B operand of WMMA f32 16x16x4 (2 VGPRs)
typedef __attribute__((ext_vector_type(8))) float v8f;  // C/D 16x16 f32 (8 VGPRs)

// ---------------------------------------------------------------- sq norms
__global__ __launch_bounds__(256)
void sqnorm_kernel(const float* __restrict__ p, float* __restrict__ pn, int count) {
    int g = blockIdx.x * blockDim.x + threadIdx.x;
    if (g >= count) return;
    float x = p[(size_t)g * 3 + 0], y = p[(size_t)g * 3 + 1], z = p[(size_t)g * 3 + 2];
    pn[g] = x * x + y * y + z * z;
}

// ---------------------------------------------------------------- KNN via WMMA
// One wave (32 lanes) handles 32 query rows of one batch.
// Cross terms q.p computed as 16x16 tiles with V_WMMA_F32_16X16X4_F32 (K=3 padded to 4).
// A 16x4 f32 layout: V0 = {K=0 | K=2}, V1 = {K=1 | K=3} across lane halves; B mirrored.
// The writer lane fuses |p|^2 into the LDS value: tile_d = |p|^2 - 2*q.p, so the
// reader (lane L owns query row L) only does d2 = |q|^2 + tile_d per candidate.
__global__ __launch_bounds__(32)
void knn_wmma_kernel(const float* __restrict__ query,  // [B,Q,3]
                     const float* __restrict__ pts,    // [B,N,3]
                     const float* __restrict__ pn,     // [B*N] |p|^2
                     int* __restrict__ out_idx,        // [B,Q,KNB] global rows into [B*N]
                     int Q, int N) {
    __shared__ float tile_d[32][17];   // padded: 17 & 64 coprime -> conflict-free row reads

    const int lane = threadIdx.x;
    const int tilesPerBatch = Q >> 5;
    const int b = blockIdx.x / tilesPerBatch;
    const int qbase = (blockIdx.x % tilesPerBatch) << 5;

    const float* qb  = query + (size_t)b * Q * 3;
    const float* pb  = pts   + (size_t)b * N * 3;
    const float* pnb = pn    + (size_t)b * N;
    const int rowbase = b * N;

    const int col = lane & 15;
    const bool hi = lane >= 16;

    // Query rows for the two A operands (rows 0-15 and 16-31 of this tile)
    const float* q0p = qb + (size_t)(qbase + col) * 3;
    const float* q1p = qb + (size_t)(qbase + 16 + col) * 3;
    float q0x = q0p[0], q0y = q0p[1], q0z = q0p[2];
    float q1x = q1p[0], q1y = q1p[1], q1z = q1p[2];

    v2f a0, a1;
    a0[0] = hi ? q0z : q0x; a0[1] = hi ? 0.f : q0y;
    a1[0] = hi ? q1z : q1x; a1[1] = hi ? 0.f : q1y;

    // Row owned by this lane (lane L -> query row qbase+L)
    float qox = hi ? q1x : q0x, qoy = hi ? q1y : q0y, qoz = hi ? q1z : q0z;
    float qn_own = qox * qox + qoy * qoy + qoz * qoz;

    float bd[KNB]; int bi[KNB];
#pragma unroll
    for (int t = 0; t < KNB; ++t) { bd[t] = 3.4e38f; bi[t] = rowbase; }

    const int ntiles = N >> 4;
    for (int j = 0; j < ntiles; ++j) {
        const int pbase = j << 4;
        const float* pp = pb + (size_t)(pbase + col) * 3;
        float px = pp[0], py = pp[1], pz = pp[2];
        float pncol = pnb[pbase + col];        // |p|^2 of this lane's column
        if (j + 1 < ntiles)
            __builtin_prefetch(pb + (size_t)(pbase + 16 + col) * 3, 0, 3);  // near prefetch

        v2f bm;
        bm[0] = hi ? pz : px; bm[1] = hi ? 0.f : py;

        v8f c0 = {};
        v8f c1 = {};
        // cross[M][N] = q[M] . p[N]  (K=3 real, lane-half K=3 slot zeroed)
        c0 = __builtin_amdgcn_wmma_f32_16x16x4_f32(false, a0, false, bm, (short)0, c0, false, false);
        c1 = __builtin_amdgcn_wmma_f32_16x16x4_f32(false, a1, false, bm, (short)0, c1, false, false);

        __syncthreads();  // previous tile fully consumed
#pragma unroll
        for (int r = 0; r < 8; ++r) {
            tile_d[(hi ? 8 : 0) + r][col]      = fmaf(-2.f, c0[r], pncol);
            tile_d[16 + (hi ? 8 : 0) + r][col] = fmaf(-2.f, c1[r], pncol);
        }
        __syncthreads();

        // lane L scans its own query row: d2 = |q|^2 + (|p|^2 - 2 q.p)
#pragma unroll
        for (int t = 0; t < 16; ++t) {
            float d = qn_own + tile_d[lane][t];
            if (d < bd[KNB - 1]) {
                // branchless compare-exchange ladder: keeps bd[] sorted ascending
                float cd = d; int ci = rowbase + pbase + t;
#pragma unroll
                for (int i = 0; i < KNB; ++i) {
                    bool lt   = cd < bd[i];
                    float nmn = fminf(cd, bd[i]);
                    float nmx = fmaxf(cd, bd[i]);
                    int   imn = lt ? ci : bi[i];
                    int   imx = lt ? bi[i] : ci;
                    bd[i] = nmn; bi[i] = imn;
                    cd = nmx;    ci = imx;
                }
            }
        }
    }

    int* orow = out_idx + ((size_t)b * Q + qbase + lane) * KNB;
#pragma unroll
    for (int t = 0; t < KNB; ++t) orow[t] = bi[t];
}

// ---------------------------------------------------------------- normals (3x3 eigh)
__device__ inline void cross3(float ax, float ay, float az, float bx, float by, float bz,
                              float& cx, float& cy, float& cz) {
    cx = ay * bz - az * by; cy = az * bx - ax * bz; cz = ax * by - ay * bx;
}

__global__ __launch_bounds__(128)
void normals_kernel(const float* __restrict__ xyz, const int* __restrict__ idx,
                    float* __restrict__ normals, float* __restrict__ lmean, int total) {
    int g = blockIdx.x * blockDim.x + threadIdx.x;
    if (g >= total) return;
    const int* ids = idx + (size_t)g * KNB;
    float cx[KNB], cy[KNB], cz[KNB];
#pragma unroll
    for (int t = 0; t < KNB; ++t) {
        int r = ids[t];
        cx[t] = xyz[(size_t)r * 3 + 0];
        cy[t] = xyz[(size_t)r * 3 + 1];
        cz[t] = xyz[(size_t)r * 3 + 2];
    }
    float mx = 0, my = 0, mz = 0;
#pragma unroll
    for (int t = 0; t < KNB; ++t) { mx += cx[t]; my += cy[t]; mz += cz[t]; }
    mx *= (1.f / KNB); my *= (1.f / KNB); mz *= (1.f / KNB);

    float a00 = 0, a01 = 0, a02 = 0, a11 = 0, a12 = 0, a22 = 0;
#pragma unroll
    for (int t = 0; t < KNB; ++t) {
        float dx = cx[t] - mx, dy = cy[t] - my, dz = cz[t] - mz;
        a00 += dx * dx; a01 += dx * dy; a02 += dx * dz;
        a11 += dy * dy; a12 += dy * dz; a22 += dz * dz;
    }
    const float inv = 1.f / (KNB - 1);
    a00 *= inv; a01 *= inv; a02 *= inv; a11 *= inv; a12 *= inv; a22 *= inv;

    // smallest eigenvalue, trigonometric method
    float p1 = a01 * a01 + a02 * a02 + a12 * a12;
    float qm = (a00 + a11 + a22) * (1.f / 3.f);
    float b00 = a00 - qm, b11 = a11 - qm, b22 = a22 - qm;
    float p2 = b00 * b00 + b11 * b11 + b22 * b22 + 2.f * p1;
    float pp = sqrtf(fmaxf(p2 * (1.f / 6.f), 0.f));
    float eig = qm;
    if (pp > 1e-20f) {
        float ip = 1.f / pp;
        float c00 = b00 * ip, c01 = a01 * ip, c02 = a02 * ip;
        float c11 = b11 * ip, c12 = a12 * ip, c22 = b22 * ip;
        float detB = c00 * (c11 * c22 - c12 * c12)
                   - c01 * (c01 * c22 - c12 * c02)
                   + c02 * (c01 * c12 - c11 * c02);
        float r = fminf(1.f, fmaxf(-1.f, 0.5f * detB));
        float phi = acosf(r) * (1.f / 3.f);
        eig = qm + 2.f * pp * cosf(phi + 2.0943951023931953f);  // + 2*pi/3 -> smallest
    }
    // eigenvector: biggest cross product of rows of (A - eig*I)
    float r0x = a00 - eig, r0y = a01, r0z = a02;
    float r1x = a01, r1y = a11 - eig, r1z = a12;
    float r2x = a02, r2y = a12, r2z = a22 - eig;
    float v0x, v0y, v0z, v1x, v1y, v1z, v2x, v2y, v2z;
    cross3(r0x, r0y, r0z, r1x, r1y, r1z, v0x, v0y, v0z);
    cross3(r0x, r0y, r0z, r2x, r2y, r2z, v1x, v1y, v1z);
    cross3(r1x, r1y, r1z, r2x, r2y, r2z, v2x, v2y, v2z);
    float n0 = v0x * v0x + v0y * v0y + v0z * v0z;
    float n1 = v1x * v1x + v1y * v1y + v1z * v1z;
    float n2 = v2x * v2x + v2y * v2y + v2z * v2z;
    float bx = v0x, by = v0y, bz = v0z, bn = n0;
    if (n1 > bn) { bx = v1x; by = v1y; bz = v1z; bn = n1; }
    if (n2 > bn) { bx = v2x; by = v2y; bz = v2z; bn = n2; }
    if (bn < 1e-30f) { bx = 0.f; by = 0.f; bz = 1.f; bn = 1.f; }
    float s = rsqrtf(bn);
    normals[(size_t)g * 3 + 0] = bx * s;
    normals[(size_t)g * 3 + 1] = by * s;
    normals[(size_t)g * 3 + 2] = bz * s;
    lmean[(size_t)g * 3 + 0] = mx;
    lmean[(size_t)g * 3 + 1] = my;
    lmean[(size_t)g * 3 + 2] = mz;
}

// ---------------------------------------------------------------- mean normal + normal-dir correction
__global__ __launch_bounds__(128)
void meannormal_kernel(const float* __restrict__ xyz, const int* __restrict__ idx,
                       const float* __restrict__ normals, const float* __restrict__ lmean,
                       float* __restrict__ mnorm, float* __restrict__ xnu,
                       float* __restrict__ pn2, int total) {
    int g = blockIdx.x * blockDim.x + threadIdx.x;
    if (g >= total) return;
    const int* ids = idx + (size_t)g * KNB;
    float sx = 0, sy = 0, sz = 0;
#pragma unroll
    for (int t = 0; t < KNB; ++t) {
        int r = ids[t];
        sx += normals[(size_t)r * 3 + 0];
        sy += normals[(size_t)r * 3 + 1];
        sz += normals[(size_t)r * 3 + 2];
    }
    sx *= (1.f / KNB); sy *= (1.f / KNB); sz *= (1.f / KNB);
    float nl = sqrtf(sx * sx + sy * sy + sz * sz);
    float is = 1.f / fmaxf(nl, EPSN);
    float nx = sx * is, ny = sy * is, nz = sz * is;
    mnorm[(size_t)g * 3 + 0] = nx;
    mnorm[(size_t)g * 3 + 1] = ny;
    mnorm[(size_t)g * 3 + 2] = nz;

    float x = xyz[(size_t)g * 3 + 0], y = xyz[(size_t)g * 3 + 1], z = xyz[(size_t)g * 3 + 2];
    float dx = x - lmean[(size_t)g * 3 + 0];
    float dy = y - lmean[(size_t)g * 3 + 1];
    float dz = z - lmean[(size_t)g * 3 + 2];
    float dp = nx * dx + ny * dy + nz * dz;
    float ux = x - nx * dp, uy = y - ny * dp, uz = z - nz * dp;
    xnu[(size_t)g * 3 + 0] = ux;
    xnu[(size_t)g * 3 + 1] = uy;
    xnu[(size_t)g * 3 + 2] = uz;
    pn2[g] = ux * ux + uy * uy + uz * uz;
}

// ---------------------------------------------------------------- FPS (one workgroup per batch)
__global__ __launch_bounds__(256)
void fps_kernel(const float* __restrict__ pts, float* __restrict__ out) {
    __shared__ float px[NPTS], py[NPTS], pz[NPTS];  // 48 KB resident point cloud
    __shared__ float rv[256];
    __shared__ int   ri[256];
    __shared__ int   s_last;
    const int tid = threadIdx.x;
    const int b = blockIdx.x;
    const float* base = pts + (size_t)b * NPTS * 3;
    float* ob = out + (size_t)b * MFPS * 3;

    for (int i = tid; i < NPTS; i += 256) {
        px[i] = base[(size_t)i * 3 + 0];
        py[i] = base[(size_t)i * 3 + 1];
        pz[i] = base[(size_t)i * 3 + 2];
    }
    if (tid == 0) s_last = 0;
    __syncthreads();
    if (tid == 0) { ob[0] = px[0]; ob[1] = py[0]; ob[2] = pz[0]; }

    float md[16];  // running min-dist for this thread's 16 strided points (registers)
#pragma unroll
    for (int t = 0; t < 16; ++t) md[t] = 3.4e38f;

    for (int s = 1; s < MFPS; ++s) {
        __syncthreads();
        int last = s_last;
        float lx = px[last], ly = py[last], lz = pz[last];
        float bv = -1.f; int bidx = 0;
#pragma unroll
        for (int t = 0; t < 16; ++t) {
            int i = t * 256 + tid;  // lane-consecutive -> bank-conflict-free
            float dx = px[i] - lx, dy = py[i] - ly, dz = pz[i] - lz;
            float d = dx * dx + dy * dy + dz * dz;
            float m = fminf(md[t], d);
            md[t] = m;
            if (m > bv) { bv = m; bidx = i; }  // strict > keeps earliest index
        }
        rv[tid] = bv; ri[tid] = bidx;
        for (int off = 128; off > 0; off >>= 1) {
            __syncthreads();
            if (tid < off) {
                float v2 = rv[tid + off]; int i2 = ri[tid + off];
                if (v2 > rv[tid] || (v2 == rv[tid] && i2 < ri[tid])) { rv[tid] = v2; ri[tid] = i2; }
            }
        }
        __syncthreads();
        if (tid == 0) {
            int w = ri[0];
            s_last = w;
            ob[(size_t)s * 3 + 0] = px[w];
            ob[(size_t)s * 3 + 1] = py[w];
            ob[(size_t)s * 3 + 2] = pz[w];
        }
    }
}

// ---------------------------------------------------------------- tangential smoothing
// out = xyz_fps - (delta - n(n.delta)) = local_mean + n*(n.delta)
__global__ __launch_bounds__(128)
void smooth_kernel(const float* __restrict__ fpspts, const float* __restrict__ xnu,
                   const float* __restrict__ mnorm, const int* __restrict__ idx,
                   float* __restrict__ out, int total) {
    int g = blockIdx.x * blockDim.x + threadIdx.x;
    if (g >= total) return;
    const int* ids = idx + (size_t)g * KNB;
    float lx = 0, ly = 0, lz = 0, sx = 0, sy = 0, sz = 0;
#pragma unroll
    for (int t = 0; t < KNB; ++t) {
        int r = ids[t];
        lx += xnu[(size_t)r * 3 + 0]; ly += xnu[(size_t)r * 3 + 1]; lz += xnu[(size_t)r * 3 + 2];
        sx += mnorm[(size_t)r * 3 + 0]; sy += mnorm[(size_t)r * 3 + 1]; sz += mnorm[(size_t)r * 3 + 2];
    }
    lx *= (1.f / KNB); ly *= (1.f / KNB); lz *= (1.f / KNB);
    sx *= (1.f / KNB); sy *= (1.f / KNB); sz *= (1.f / KNB);
    float nl = sqrtf(sx * sx + sy * sy + sz * sz);
    float is = 1.f / fmaxf(nl, EPSN);
    float nx = sx * is, ny = sy * is, nz = sz * is;

    float fx = fpspts[(size_t)g * 3 + 0], fy = fpspts[(size_t)g * 3 + 1], fz = fpspts[(size_t)g * 3 + 2];
    float dx = fx - lx, dy = fy - ly, dz = fz - lz;
    float dp = nx * dx + ny * dy + nz * dz;
    out[(size_t)g * 3 + 0] = lx + nx * dp;
    out[(size_t)g * 3 + 1] = ly + ny * dp;
    out[(size_t)g * 3 + 2] = lz + nz * dp;
}

// ---------------------------------------------------------------- launch
extern "C" void kernel_launch(void* const* d_in, const int* in_sizes, int n_in,
                              void* d_out, int out_size, void* d_ws, size_t ws_size,
                              hipStream_t stream) {
    const float* xyz = (const float*)d_in[0];
    const int N = NPTS;
    const int B = in_sizes[0] / (N * 3);          // 16 for the reference setup
    const int total = B * N;
    const int totalM = B * MFPS;

    // carve scratch (~8 MB total) out of d_ws
    size_t off = 0;
    auto carve = [&](size_t bytes) -> void* {
        void* p = (char*)d_ws + off;
        off += (bytes + 255) & ~(size_t)255;
        return p;
    };
    float* pn1     = (float*)carve(sizeof(float) * (size_t)total);
    int*   idx1    = (int*)  carve(sizeof(int)   * (size_t)total * KNB);
    float* normals = (float*)carve(sizeof(float) * (size_t)total * 3);
    float* lmean   = (float*)carve(sizeof(float) * (size_t)total * 3);
    float* mnorm   = (float*)carve(sizeof(float) * (size_t)total * 3);
    float* xnu     = (float*)carve(sizeof(float) * (size_t)total * 3);
    float* pn2     = (float*)carve(sizeof(float) * (size_t)total);
    float* fpsbuf  = (float*)carve(sizeof(float) * (size_t)totalM * 3);
    int*   idx2    = (int*)  carve(sizeof(int)   * (size_t)totalM * KNB);
    (void)ws_size; (void)n_in; (void)out_size;

    sqnorm_kernel<<<(total + 255) / 256, 256, 0, stream>>>(xyz, pn1, total);
    knn_wmma_kernel<<<total / 32, 32, 0, stream>>>(xyz, xyz, pn1, idx1, N, N);
    normals_kernel<<<(total + 127) / 128, 128, 0, stream>>>(xyz, idx1, normals, lmean, total);
    meannormal_kernel<<<(total + 127) / 128, 128, 0, stream>>>(xyz, idx1, normals, lmean,
                                                               mnorm, xnu, pn2, total);
    fps_kernel<<<B, 256, 0, stream>>>(xnu, fpsbuf);
    knn_wmma_kernel<<<totalM / 32, 32, 0, stream>>>(fpsbuf, xnu, pn2, idx2, MFPS, N);
    smooth_kernel<<<(totalM + 127) / 128, 128, 0, stream>>>(fpsbuf, xnu, mnorm, idx2,
                                                            (float*)d_out, totalM);
}